// SpikeEncoder_22127671509476
// MI455X (gfx1250) — compile-verified
//
#include <hip/hip_runtime.h>
#include <hip/hip_bf16.h>
#include <stdint.h>

// Problem constants (match reference setup_inputs)
#define B_      8
#define S_      2048
#define D_      1536
#define NTOK    (B_ * S_)            // 16384 tokens / rows
#define NELEM   ((size_t)NTOK * D_)  // 25,165,824 elements per output tensor

#define LEAK_F      0.95f
#define THRESH_F    1.0f
#define CLAMP_F     3.0f
#define LN_EPS_F    1e-5f

// LIF-scan tiling
#define CH   64                      // channels per block (one slice of D)
#define TT   32                      // timesteps per TDM tile
#define NT   (S_ / TT)               // 64 tiles along time

typedef unsigned int u32x4 __attribute__((ext_vector_type(4)));
typedef int          i32x4 __attribute__((ext_vector_type(4)));
typedef int          i32x8 __attribute__((ext_vector_type(8)));

// ---------------------------------------------------------------------------
// TDM helpers: build D# descriptor groups per CDNA5 ISA.
// Group0: [1:0]count=1 | lds_addr[63:32] | global_addr[120:64] | type=2[127:126]
// Group1: data_size=4B, tensor_dim0=D_, tensor_dim1=NTOK, tile=CHxTT,
//         tensor_dim0_stride=D_ (elements). 2-D tile: groups 2/3 zero.
// ---------------------------------------------------------------------------
__device__ __forceinline__ unsigned lds_byte_addr(const void* p) {
    // Generic pointer to LDS: low 32 bits are the wave-relative LDS address.
    return (unsigned)(uint64_t)(uintptr_t)p;
}

__device__ __forceinline__ void make_desc(uint64_t gaddr, unsigned lds,
                                          u32x4& g0, i32x8& g1) {
    g0[0] = 1u;                                        // count=1, user mode
    g0[1] = lds;                                       // lds_addr (bytes)
    g0[2] = (unsigned)(gaddr & 0xffffffffu);           // global_addr lo
    g0[3] = (unsigned)((gaddr >> 32) & 0x01ffffffu)    // global_addr hi
          | (2u << 30);                                // type = 2 ("image")
    g1[0] = (int)(2u << 16);                           // data_size code 2 = 4B
    g1[1] = (int)((unsigned)D_ << 16);                 // tensor_dim0[15:0] @ w1[31:16]
    g1[2] = (int)((unsigned)(NTOK & 0xffff) << 16);    // tensor_dim1[15:0] @ w2[31:16]
    g1[3] = (int)(((unsigned)CH) << 16)                // tile_dim0 @ w3[31:16]
          | (int)((unsigned)NTOK >> 16);               // tensor_dim1[31:16] @ w3[15:0]
    g1[4] = TT;                                        // tile_dim1 ; tile_dim2=0
    g1[5] = D_;                                        // tensor_dim0_stride lo32
    g1[6] = 0;                                         // stride hi / dim1_stride lo
    g1[7] = 0;
}

__device__ __forceinline__ void tdm_load(uint64_t gaddr, unsigned lds) {
    u32x4 g0; i32x8 g1;
    make_desc(gaddr, lds, g0, g1);
    i32x4 z4 = (i32x4)0;
#if __clang_major__ >= 23
    i32x8 z8 = (i32x8)0;
    __builtin_amdgcn_tensor_load_to_lds(g0, g1, z4, z4, z8, 0);
#else
    __builtin_amdgcn_tensor_load_to_lds(g0, g1, z4, z4, 0);
#endif
}

__device__ __forceinline__ void tdm_store(uint64_t gaddr, unsigned lds) {
    u32x4 g0; i32x8 g1;
    make_desc(gaddr, lds, g0, g1);
    i32x4 z4 = (i32x4)0;
#if __clang_major__ >= 23
    i32x8 z8 = (i32x8)0;
    __builtin_amdgcn_tensor_store_from_lds(g0, g1, z4, z4, z8, 0);
#else
    __builtin_amdgcn_tensor_store_from_lds(g0, g1, z4, z4, 0);
#endif
}

// ---------------------------------------------------------------------------
// Kernel 1: embedding gather + LayerNorm. One wave32 per token.
// Lane L holds columns {j*128 + L*4 .. +3}, j = 0..11 (48 floats / lane).
// ---------------------------------------------------------------------------
__device__ __forceinline__ float wave_sum(float v) {
#pragma unroll
    for (int off = 16; off > 0; off >>= 1)
        v += __shfl_xor(v, off, 32);
    return v;
}

__global__ __launch_bounds__(256) void gather_ln_kernel(
    const int*   __restrict__ token_ids,
    const float* __restrict__ emb,
    const float* __restrict__ gamma,
    const float* __restrict__ beta,
    float*       __restrict__ x_out)
{
    const int lane  = threadIdx.x & 31;
    const int wave  = threadIdx.x >> 5;
    const int token = blockIdx.x * 8 + wave;

    const int tok = token_ids[token];
    const float* __restrict__ row = emb + (size_t)tok * D_;

    float4 v[12];
#pragma unroll
    for (int j = 0; j < 12; ++j) {
        const int col = j * 128 + lane * 4;
        v[j] = *(const float4*)(row + col);
    }

    float s = 0.0f;
#pragma unroll
    for (int j = 0; j < 12; ++j)
        s += (v[j].x + v[j].y) + (v[j].z + v[j].w);
    s = wave_sum(s);
    const float mu = s * (1.0f / (float)D_);

    float q = 0.0f;
#pragma unroll
    for (int j = 0; j < 12; ++j) {
        const float a = v[j].x - mu, b = v[j].y - mu;
        const float c = v[j].z - mu, d = v[j].w - mu;
        q += (a * a + b * b) + (c * c + d * d);
    }
    q = wave_sum(q);
    const float rstd = rsqrtf(q * (1.0f / (float)D_) + LN_EPS_F);

    float* __restrict__ out = x_out + (size_t)token * D_;
#pragma unroll
    for (int j = 0; j < 12; ++j) {
        const int col = j * 128 + lane * 4;
        const float4 g  = *(const float4*)(gamma + col);
        const float4 bb = *(const float4*)(beta + col);
        float4 y;
        y.x = (v[j].x - mu) * rstd * g.x + bb.x;
        y.y = (v[j].y - mu) * rstd * g.y + bb.y;
        y.z = (v[j].z - mu) * rstd * g.z + bb.z;
        y.w = (v[j].w - mu) * rstd * g.w + bb.w;
        *(float4*)(out + col) = y;
    }
}

// ---------------------------------------------------------------------------
// Kernel 2: LIF scan over time, TDM-pipelined.
// Block = 64 threads (2 waves) owns one (batch b, 64-channel slice d0..d0+63).
// Wave 0 drives the Tensor Data Mover: double-buffered 32x64 f32 input tiles
// of x and output tiles of spikes. TDM ops from one wave complete in order,
// so s_wait_tensorcnt(2) after issuing load(i+1) guarantees load(i) AND the
// store reusing outbuf are done.  Issue order: L0 L1 [S0 L2] [S1 L3] ...
// ---------------------------------------------------------------------------
__global__ __launch_bounds__(64) void lif_scan_kernel(
    const float* __restrict__ x,       // [B,S,D] layernormed stream
    float*       __restrict__ spikes)  // [B,S,D] 0/1 output
{
    __shared__ float inbuf [2][TT * CH];
    __shared__ float outbuf[2][TT * CH];

    const int blocksPerB = D_ / CH;                 // 24
    const int b   = blockIdx.x / blocksPerB;
    const int d0  = (blockIdx.x % blocksPerB) * CH;
    const int tid = threadIdx.x;

    const uint64_t x_base  = (uint64_t)(uintptr_t)x;
    const uint64_t sp_base = (uint64_t)(uintptr_t)spikes;
    const uint64_t row0    = (uint64_t)(b * S_) * D_ + (uint64_t)d0;

    float v = 0.0f;

    if (tid < 32) {  // wave 0 only (uniform branch; wave 1 skips via execz)
        tdm_load(x_base + 4ull * (row0 + 0ull * TT * D_), lds_byte_addr(&inbuf[0][0]));
        tdm_load(x_base + 4ull * (row0 + 1ull * TT * D_), lds_byte_addr(&inbuf[1][0]));
    }

    for (int i = 0; i < NT; ++i) {
        const int cur = i & 1;

        if (tid < 32) {
            if (i == 0) {
                __builtin_amdgcn_s_wait_tensorcnt(1);   // only L1 may remain
            } else if (i + 1 < NT) {
                __builtin_amdgcn_s_wait_tensorcnt(2);   // S(i-1), L(i+1) may remain
            } else {
                __builtin_amdgcn_s_wait_tensorcnt(1);   // tail: S(i-1) may remain
            }
        }
        __syncthreads();  // tile i resident in inbuf[cur]; outbuf[cur] reusable

#pragma unroll
        for (int r = 0; r < TT; ++r) {
            const float xv = inbuf[cur][r * CH + tid];
            v = v * LEAK_F + xv;
            v = fminf(fmaxf(v, -CLAMP_F), CLAMP_F);
            const float spk = (v >= THRESH_F) ? 1.0f : 0.0f;
            outbuf[cur][r * CH + tid] = spk;
            v = v * (1.0f - spk);
        }
        __syncthreads();  // all lanes done with inbuf[cur] / outbuf[cur] writes

        if (tid < 32) {
            const uint64_t off = row0 + (uint64_t)i * TT * D_;
            tdm_store(sp_base + 4ull * off, lds_byte_addr(&outbuf[cur][0]));
            if (i + 2 < NT) {
                const uint64_t noff = row0 + (uint64_t)(i + 2) * TT * D_;
                tdm_load(x_base + 4ull * noff, lds_byte_addr(&inbuf[cur][0]));
            }
        }
    }
    // s_endpgm performs an implicit wait-idle, draining the final TDM store.
}

// ---------------------------------------------------------------------------
// Host-side launch
// ---------------------------------------------------------------------------
extern "C" void kernel_launch(void* const* d_in, const int* in_sizes, int n_in,
                              void* d_out, int out_size, void* d_ws, size_t ws_size,
                              hipStream_t stream) {
    const int*   token_ids = (const int*)  d_in[0];
    const float* emb       = (const float*)d_in[1];
    const float* gamma     = (const float*)d_in[2];
    const float* beta      = (const float*)d_in[3];

    float* spikes = (float*)d_out;          // output 0: [B,S,D]
    float* x      = spikes + NELEM;         // output 1: [B,S,D] (also intermediate)

    (void)in_sizes; (void)n_in; (void)out_size; (void)d_ws; (void)ws_size;

    // Phase 1: gather + layernorm -> x (8 tokens per 256-thread block)
    gather_ln_kernel<<<NTOK / 8, 256, 0, stream>>>(token_ids, emb, gamma, beta, x);

    // Phase 2: sequential LIF scan over S, TDM-tiled
    lif_scan_kernel<<<B_ * (D_ / CH), 64, 0, stream>>>(x, spikes);
}